// SymmetricSplatting_77884936946166
// MI455X (gfx1250) — compile-verified
//
#include <hip/hip_runtime.h>
#include <hip/hip_bf16.h>

// Problem constants from the reference: B=4, C=16, H=512, W=512.
#define B_  4
#define C_  16
#define H_  512
#define W_  512
#define HW_ (H_ * W_)

// ---------------------------------------------------------------------------
// Kernel 1: zero the accumulators with b128 stores.
//   d_out : [B,16,H,W] value accumulator  (nOut4  float4 elements)
//   d_ws  : [B,H,W]    norm accumulator   (nNorm4 float4 elements)
// Both counts are multiples of 4, so no tail handling needed.
// ---------------------------------------------------------------------------
__global__ void __launch_bounds__(256)
splat_zero_kernel(float4* __restrict__ out, float4* __restrict__ norm,
                  int nOut4, int nNorm4) {
    const int i = blockIdx.x * blockDim.x + threadIdx.x;
    const float4 z = make_float4(0.f, 0.f, 0.f, 0.f);
    if (i < nOut4)  out[i]  = z;   // global_store_b128
    if (i < nNorm4) norm[i] = z;   // global_store_b128
}

// ---------------------------------------------------------------------------
// Kernel 2: one splat direction. Each thread owns one source pixel (b,y,x):
//   ms = exp(metric) * dirScale   (dirScale = 1-alpha fwd, alpha bwd)
//   for c in 0..15: atomically scatter tensor[c]*ms into the 4 bilinear
//   corners of the shared accumulator; then scatter ms into the norm plane.
// Loads are fully coalesced (lane index == x). Atomics are return-less
// (global_atomic_add_f32 -> STOREcnt, fire-and-forget, resolved in L2 since
// the whole 71 MB accumulator fits in the 192 MB L2). global_prefetch_b8
// pulls the next channel's cacheline while this channel's atomics are in
// flight.
// ---------------------------------------------------------------------------
__global__ void __launch_bounds__(256)
splat_scatter_kernel(const float* __restrict__ tensor,   // [B,16,H,W]
                     const float* __restrict__ flow,     // [B,2,H,W]
                     const float* __restrict__ metric,   // [B,1,H,W]
                     const float* __restrict__ t,        // [B]
                     const int*   __restrict__ Nptr,     // [1]
                     float* __restrict__ acc,            // [B,16,H,W]
                     float* __restrict__ norm,           // [B,H,W]
                     int isForward) {
    const int p = blockIdx.x * blockDim.x + threadIdx.x;   // pixel in plane
    const int b = blockIdx.y;
    if (p >= HW_) return;

    const int x = p & (W_ - 1);
    const int y = p >> 9;                                  // W == 512

    const float alpha    = t[b] / (float)Nptr[0];
    const float dirScale = isForward ? (1.0f - alpha) : alpha;

    const float* __restrict__ fb = flow + (size_t)b * 2 * HW_;
    const float dx = fb[p];
    const float dy = fb[HW_ + p];

    const float fx  = (float)x + dx;
    const float fy  = (float)y + dy;
    const float x0f = floorf(fx);
    const float y0f = floorf(fy);
    const int   x0  = (int)x0f;
    const int   y0  = (int)y0f;

    const float wx1 = fx - x0f, wx0 = 1.0f - wx1;
    const float wy1 = fy - y0f, wy0 = 1.0f - wy1;

    // Corner validity (reference drops OOB corners).
    const bool vx0 = (x0 >= 0)     && (x0 < W_);
    const bool vx1 = (x0 + 1 >= 0) && (x0 + 1 < W_);
    const bool vy0 = (y0 >= 0)     && (y0 < H_);
    const bool vy1 = (y0 + 1 >= 0) && (y0 + 1 < H_);

    const bool v00 = vx0 && vy0, v01 = vx1 && vy0;
    const bool v10 = vx0 && vy1, v11 = vx1 && vy1;

    const int i00 = y0 * W_ + x0;
    const int i01 = i00 + 1;
    const int i10 = i00 + W_;
    const int i11 = i10 + 1;

    const float w00 = wx0 * wy0, w01 = wx1 * wy0;
    const float w10 = wx0 * wy1, w11 = wx1 * wy1;

    // exp(metric) folded with the per-batch direction scale: one per-pixel
    // multiplier covers the value channels AND the normalization channel.
    const float ms = __expf(metric[(size_t)b * HW_ + p]) * dirScale;

    const float* __restrict__ tb   = tensor + (size_t)b * C_ * HW_;
    float*       __restrict__ accb = acc    + (size_t)b * C_ * HW_;

#pragma unroll 4
    for (int c = 0; c < C_; ++c) {
        const float v = tb[(size_t)c * HW_ + p] * ms;
        if (c + 1 < C_) {
            // gfx1250: lowers to global_prefetch_b8.
            __builtin_prefetch(&tb[(size_t)(c + 1) * HW_ + p], 0, 0);
        }
        float* plane = accb + (size_t)c * HW_;
        if (v00) atomicAdd(plane + i00, v * w00);
        if (v01) atomicAdd(plane + i01, v * w01);
        if (v10) atomicAdd(plane + i10, v * w10);
        if (v11) atomicAdd(plane + i11, v * w11);
    }

    float* nplane = norm + (size_t)b * HW_;
    if (v00) atomicAdd(nplane + i00, ms * w00);
    if (v01) atomicAdd(nplane + i01, ms * w01);
    if (v10) atomicAdd(nplane + i10, ms * w10);
    if (v11) atomicAdd(nplane + i11, ms * w11);
}

// ---------------------------------------------------------------------------
// Kernel 3: normalize 4 adjacent pixels per thread with b128 traffic.
// out[b,c,p] /= (norm==0 ? 1 : norm), elementwise over the float4.
// ---------------------------------------------------------------------------
__global__ void __launch_bounds__(256)
splat_normalize_kernel(float4* __restrict__ out,          // [B,16,HW/4]
                       const float4* __restrict__ norm) { // [B,HW/4]
    const int q = blockIdx.x * blockDim.x + threadIdx.x;  // float4 idx in plane
    const int b = blockIdx.y;
    const int QW = HW_ / 4;
    if (q >= QW) return;

    const float4 n = norm[(size_t)b * QW + q];            // global_load_b128
    float4 inv;
    inv.x = 1.0f / ((n.x == 0.0f) ? 1.0f : n.x);
    inv.y = 1.0f / ((n.y == 0.0f) ? 1.0f : n.y);
    inv.z = 1.0f / ((n.z == 0.0f) ? 1.0f : n.z);
    inv.w = 1.0f / ((n.w == 0.0f) ? 1.0f : n.w);

    float4* ob = out + (size_t)b * C_ * QW;
#pragma unroll
    for (int c = 0; c < C_; ++c) {
        float4 v = ob[(size_t)c * QW + q];                // global_load_b128
        v.x *= inv.x; v.y *= inv.y; v.z *= inv.z; v.w *= inv.w;
        ob[(size_t)c * QW + q] = v;                       // global_store_b128
    }
}

// ---------------------------------------------------------------------------
extern "C" void kernel_launch(void* const* d_in, const int* in_sizes, int n_in,
                              void* d_out, int out_size, void* d_ws, size_t ws_size,
                              hipStream_t stream) {
    const float* ftensor = (const float*)d_in[0];
    const float* fflow   = (const float*)d_in[1];
    const float* fmetric = (const float*)d_in[2];
    const float* btensor = (const float*)d_in[3];
    const float* bflow   = (const float*)d_in[4];
    const float* bmetric = (const float*)d_in[5];
    const float* t       = (const float*)d_in[6];
    const int*   N       = (const int*)d_in[7];

    float* out  = (float*)d_out;            // [B,16,H,W] accumulator -> final
    float* norm = (float*)d_ws;             // [B,H,W] normalization accumulator

    const int nOut4  = (B_ * C_ * HW_) / 4; // 4,194,304 float4
    const int nNorm4 = (B_ * HW_) / 4;      //   262,144 float4

    splat_zero_kernel<<<(nOut4 + 255) / 256, 256, 0, stream>>>(
        (float4*)out, (float4*)norm, nOut4, nNorm4);

    dim3 grid(HW_ / 256, B_);
    splat_scatter_kernel<<<grid, 256, 0, stream>>>(ftensor, fflow, fmetric, t, N,
                                                   out, norm, /*isForward=*/1);
    splat_scatter_kernel<<<grid, 256, 0, stream>>>(btensor, bflow, bmetric, t, N,
                                                   out, norm, /*isForward=*/0);

    dim3 ngrid((HW_ / 4 + 255) / 256, B_);
    splat_normalize_kernel<<<ngrid, 256, 0, stream>>>((float4*)out,
                                                      (const float4*)norm);
}